// InfGIN_88416196755456
// MI455X (gfx1250) — compile-verified
//
#include <hip/hip_runtime.h>
#include <hip/hip_bf16.h>

// ---------------- problem constants (match reference) ----------------
#define NNODES   100000
#define NEDGES   1200000
#define DIMV     64
#define NGRAPHS  128
#define NLAYERS  5

typedef __attribute__((ext_vector_type(2))) float v2f;
typedef __attribute__((ext_vector_type(8))) float v8f;

#define LDS_STRIDE 66   // 64 + 2 pad: strided A-fragment reads hit distinct banks

// ---------------------------------------------------------------------
// acc[i] = h[i]   (initialize z = h before edge scatter-add)
// ---------------------------------------------------------------------
__global__ __launch_bounds__(256)
void gin_copy_f4(const float4* __restrict__ in, float4* __restrict__ out, int n4) {
    int i = blockIdx.x * 256 + threadIdx.x;
    if (i < n4) out[i] = in[i];
}

// ---------------------------------------------------------------------
// acc[dst[e]] += h[src[e]]  — 16 threads per edge, float4 gather + 4 f32 atomics
// ei layout: [0,NEDGES) = src, [NEDGES,2*NEDGES) = dst
// ---------------------------------------------------------------------
__global__ __launch_bounds__(256)
void gin_edge_scatter(const float* __restrict__ h, const int* __restrict__ ei,
                      float* __restrict__ acc) {
    unsigned idx = blockIdx.x * 256u + threadIdx.x;   // exact: NEDGES*16 threads
    int e = idx >> 4;
    int q = idx & 15;
    int s = ei[e];
    int d = ei[NEDGES + e];
    float4 v = ((const float4*)(h + (size_t)s * DIMV))[q];
    float* o = acc + (size_t)d * DIMV + q * 4;
    atomicAdd(o + 0, v.x);
    atomicAdd(o + 1, v.y);
    atomicAdd(o + 2, v.z);
    atomicAdd(o + 3, v.w);
}

// ---------------------------------------------------------------------
// Fused GIN MLP:  out = relu( relu((z@W1 + b1)*scale + beta) @ W2 + b2 )
// Block = 256 threads = 8 waves; block tile = 32 nodes x 64 dims.
// wave -> (mtile = wave>>2 in 0..1, ntile = wave&3 in 0..3), 16x16 C tile per wave.
// Two V_WMMA_F32_16X16X4_F32 passes (16 K-steps each), intermediate via LDS.
// ---------------------------------------------------------------------
__global__ __launch_bounds__(256)
void gin_mlp(const float* __restrict__ zin,
             const float* __restrict__ W1, const float* __restrict__ b1,
             const float* __restrict__ gamma, const float* __restrict__ beta,
             const float* __restrict__ W2, const float* __restrict__ b2,
             float* __restrict__ out) {
    __shared__ float ldsA[32 * LDS_STRIDE];
    __shared__ float ldsM[32 * LDS_STRIDE];

    const int tid   = threadIdx.x;
    const int lane  = tid & 31;
    const int wave  = tid >> 5;
    const int mtile = wave >> 2;       // 0..1
    const int ntile = wave & 3;        // 0..3
    const int nodeBase = blockIdx.x * 32;

    // ---- stage 32x64 input tile into LDS (coalesced float4 global loads)
    for (int i = tid; i < 32 * 16; i += 256) {
        int node = i >> 4;             // 0..31
        int q    = i & 15;             // float4 index within row
        float4 v = ((const float4*)(zin + (size_t)(nodeBase + node) * DIMV))[q];
        float2* p = (float2*)&ldsA[node * LDS_STRIDE + q * 4]; // 8B aligned (stride even)
        p[0] = make_float2(v.x, v.y);
        p[1] = make_float2(v.z, v.w);
    }
    __syncthreads();

    const int col   = ntile * 16 + (lane & 15);     // output column (N)
    const int khalf = (lane >> 4) << 1;             // 0 (lanes 0-15) or 2 (lanes 16-31)
    const int arow  = mtile * 16 + (lane & 15);     // A-matrix row (M)
    const int rbase = (lane >> 4) * 8;              // C row base per half-wave

    // ================= GEMM1: z @ W1 + b1, BN, ReLU -> ldsM =================
    {
        float bias = b1[col];
        v8f c;
        #pragma unroll
        for (int r = 0; r < 8; ++r) c[r] = bias;

        #pragma unroll
        for (int k0 = 0; k0 < DIMV; k0 += 4) {
            v2f a, b;
            a.x = ldsA[arow * LDS_STRIDE + k0 + khalf];
            a.y = ldsA[arow * LDS_STRIDE + k0 + khalf + 1];
            b.x = W1[(k0 + khalf) * DIMV + col];
            b.y = W1[(k0 + khalf + 1) * DIMV + col];
            c = __builtin_amdgcn_wmma_f32_16x16x4_f32(
                    false, a, false, b, (short)0, c, false, false);
        }

        float sc = gamma[col] * rsqrtf(1.0f + 1e-5f);
        float be = beta[col];
        #pragma unroll
        for (int r = 0; r < 8; ++r) {
            float v = fmaxf(c[r] * sc + be, 0.0f);
            ldsM[(mtile * 16 + rbase + r) * LDS_STRIDE + ntile * 16 + (lane & 15)] = v;
        }
    }
    __syncthreads();

    // ================= GEMM2: mid @ W2 + b2, ReLU -> global =================
    {
        float bias = b2[col];
        v8f c;
        #pragma unroll
        for (int r = 0; r < 8; ++r) c[r] = bias;

        #pragma unroll
        for (int k0 = 0; k0 < DIMV; k0 += 4) {
            v2f a, b;
            a.x = ldsM[arow * LDS_STRIDE + k0 + khalf];
            a.y = ldsM[arow * LDS_STRIDE + k0 + khalf + 1];
            b.x = W2[(k0 + khalf) * DIMV + col];
            b.y = W2[(k0 + khalf + 1) * DIMV + col];
            c = __builtin_amdgcn_wmma_f32_16x16x4_f32(
                    false, a, false, b, (short)0, c, false, false);
        }

        #pragma unroll
        for (int r = 0; r < 8; ++r) {
            float v = fmaxf(c[r], 0.0f);
            out[(size_t)(nodeBase + mtile * 16 + rbase + r) * DIMV + col] = v;
        }
    }
}

// ---------------------------------------------------------------------
// pooled[batch[n]] += h[n]  — 16 threads per node, 4 f32 atomics each
// ---------------------------------------------------------------------
__global__ __launch_bounds__(256)
void gin_pool_scatter(const float* __restrict__ h, const int* __restrict__ batch,
                      float* __restrict__ pooled) {
    unsigned idx = blockIdx.x * 256u + threadIdx.x;   // exact: NNODES*16 threads
    int node = idx >> 4;
    int q    = idx & 15;
    int g = batch[node];
    float4 v = ((const float4*)(h + (size_t)node * DIMV))[q];
    float* o = pooled + (size_t)g * DIMV + q * 4;
    atomicAdd(o + 0, v.x);
    atomicAdd(o + 1, v.y);
    atomicAdd(o + 2, v.z);
    atomicAdd(o + 3, v.w);
}

// ---------------------------------------------------------------------
// out = relu(pooled @ lin1_w + lin1_b)   [128x64 @ 64x64]
// grid = 8 blocks x 128 threads (4 waves, one 16-col tile per wave)
// ---------------------------------------------------------------------
__global__ __launch_bounds__(128)
void gin_final_gemm(const float* __restrict__ pooled, const float* __restrict__ W,
                    const float* __restrict__ bvec, float* __restrict__ out) {
    const int tid   = threadIdx.x;
    const int lane  = tid & 31;
    const int wave  = tid >> 5;        // ntile 0..3
    const int g0    = blockIdx.x * 16; // graph-row tile
    const int col   = wave * 16 + (lane & 15);
    const int khalf = (lane >> 4) << 1;
    const int arow  = g0 + (lane & 15);
    const int rbase = (lane >> 4) * 8;

    float bias = bvec[col];
    v8f c;
    #pragma unroll
    for (int r = 0; r < 8; ++r) c[r] = bias;

    #pragma unroll
    for (int k0 = 0; k0 < DIMV; k0 += 4) {
        v2f a, b;
        a.x = pooled[arow * DIMV + k0 + khalf];
        a.y = pooled[arow * DIMV + k0 + khalf + 1];
        b.x = W[(k0 + khalf) * DIMV + col];
        b.y = W[(k0 + khalf + 1) * DIMV + col];
        c = __builtin_amdgcn_wmma_f32_16x16x4_f32(
                false, a, false, b, (short)0, c, false, false);
    }

    #pragma unroll
    for (int r = 0; r < 8; ++r) {
        out[(size_t)(g0 + rbase + r) * DIMV + col] = fmaxf(c[r], 0.0f);
    }
}

// ---------------------------------------------------------------------
extern "C" void kernel_launch(void* const* d_in, const int* in_sizes, int n_in,
                              void* d_out, int out_size, void* d_ws, size_t ws_size,
                              hipStream_t stream) {
    const float* x      = (const float*)d_in[0];
    const int*   ei     = (const int*)d_in[1];   // (2, NEDGES) row-major
    const int*   batch  = (const int*)d_in[2];
    const float* W1s    = (const float*)d_in[3]; // (5,64,64)
    const float* b1s    = (const float*)d_in[4]; // (5,64)
    const float* gammas = (const float*)d_in[5];
    const float* betas  = (const float*)d_in[6];
    const float* W2s    = (const float*)d_in[7];
    const float* b2s    = (const float*)d_in[8];
    const float* lin1_w = (const float*)d_in[9];
    const float* lin1_b = (const float*)d_in[10];
    float* out = (float*)d_out;

    // workspace: bufA (h, 25.6MB) | bufB (acc, 25.6MB) | pooled (32KB)
    const size_t hbytes = (size_t)NNODES * DIMV * sizeof(float);
    char* ws = (char*)d_ws;
    float* bufA   = (float*)ws;
    float* bufB   = (float*)(ws + hbytes);
    float* pooled = (float*)(ws + 2 * hbytes);

    const int n4 = NNODES * DIMV / 4;                 // 1,600,000 float4
    const int copyBlocks = n4 / 256;                  // 6250
    const int edgeBlocks = (NEDGES * 16) / 256;       // 75000
    const int mlpBlocks  = NNODES / 32;               // 3125
    const int poolBlocks = (NNODES * 16) / 256;       // 6250

    const float* h = x;
    for (int l = 0; l < NLAYERS; ++l) {
        gin_copy_f4<<<copyBlocks, 256, 0, stream>>>((const float4*)h, (float4*)bufB, n4);
        gin_edge_scatter<<<edgeBlocks, 256, 0, stream>>>(h, ei, bufB);
        gin_mlp<<<mlpBlocks, 256, 0, stream>>>(bufB,
                                               W1s + (size_t)l * DIMV * DIMV,
                                               b1s + (size_t)l * DIMV,
                                               gammas + (size_t)l * DIMV,
                                               betas + (size_t)l * DIMV,
                                               W2s + (size_t)l * DIMV * DIMV,
                                               b2s + (size_t)l * DIMV,
                                               bufA);
        h = bufA;   // acc (bufB) is the only MLP input, so bufA can be reused in place
    }

    hipMemsetAsync(pooled, 0, (size_t)NGRAPHS * DIMV * sizeof(float), stream);
    gin_pool_scatter<<<poolBlocks, 256, 0, stream>>>(h, batch, pooled);
    gin_final_gemm<<<8, 128, 0, stream>>>(pooled, lin1_w, lin1_b, out);
}